// Hyper_SAGNN_40355512713729
// MI455X (gfx1250) — compile-verified
//
#include <hip/hip_runtime.h>
#include <hip/hip_bf16.h>

typedef __attribute__((ext_vector_type(2))) float v2f;
typedef __attribute__((ext_vector_type(8))) float v8f;

#define LDS_STRIDE 516   // 512 + 4 pad floats: rows shift 4 LDS banks -> conflict-free

__global__ __launch_bounds__(256)
void hsagnn_fused_kernel(const int*   __restrict__ nodes,      // (B,5)
                         const int*   __restrict__ mask_col,   // (N*32)
                         const float* __restrict__ mask_val,   // (N*32)
                         const int*   __restrict__ uniq,       // (U)
                         const float* __restrict__ embed,      // (V,256)
                         const float* __restrict__ Wm,         // (256,512)
                         const float* __restrict__ bias,       // (256)
                         float*       __restrict__ out)        // (B,5,256)
{
    __shared__ float sC[16 * LDS_STRIDE];   // 16 x 512 combined tile (padded)

    const int blk  = blockIdx.x;            // 0..1023, owns 16 segments / 4 batches
    const int tid  = threadIdx.x;
    const int w    = tid >> 5;              // wave 0..7
    const int lane = tid & 31;

    // ---------------- Phase 1: neighbor aggregation + self feats -> LDS ------
    // Each wave handles 2 segments; each lane accumulates 8 floats of the row.
    for (int ss = 0; ss < 2; ++ss) {
        const int  seg  = w * 2 + ss;               // 0..15 local row
        const int  gseg = blk * 16 + seg;           // global segment (0..N-1)
        const long base = (long)gseg * 32;

        float a0x=0.f,a0y=0.f,a0z=0.f,a0w=0.f;
        float a1x=0.f,a1y=0.f,a1z=0.f,a1w=0.f;

        for (int j = 0; j < 32; ++j) {
            const int   c = mask_col[base + j];
            const int   r = uniq[c];
            const float v = mask_val[base + j];
            const float4* row = (const float4*)(embed + (size_t)r * 256);
            const float4 x0 = row[lane * 2];
            const float4 x1 = row[lane * 2 + 1];
            a0x = fmaf(v, x0.x, a0x); a0y = fmaf(v, x0.y, a0y);
            a0z = fmaf(v, x0.z, a0z); a0w = fmaf(v, x0.w, a0w);
            a1x = fmaf(v, x1.x, a1x); a1y = fmaf(v, x1.y, a1y);
            a1z = fmaf(v, x1.z, a1z); a1w = fmaf(v, x1.w, a1w);
        }
        float* dst = &sC[seg * LDS_STRIDE + lane * 8];
        ((float4*)dst)[0] = make_float4(a0x, a0y, a0z, a0w);
        ((float4*)dst)[1] = make_float4(a1x, a1y, a1z, a1w);

        // self feats -> columns 256..511
        const int batch = gseg >> 2;
        const int t     = gseg & 3;
        const int node  = nodes[batch * 5 + t + 1];
        const float4* srow = (const float4*)(embed + (size_t)node * 256);
        const float4 s0 = srow[lane * 2];
        const float4 s1 = srow[lane * 2 + 1];
        float* dst2 = &sC[seg * LDS_STRIDE + 256 + lane * 8];
        ((float4*)dst2)[0] = s0;
        ((float4*)dst2)[1] = s1;
    }

    // cell feats: out[batch][0][:] = embed[nodes[batch][0]] (waves 0..3, 1 batch each)
    if (w < 4) {
        const int batch = blk * 4 + w;
        const int node  = nodes[batch * 5];
        const float4* crow = (const float4*)(embed + (size_t)node * 256);
        const float4 c0 = crow[lane * 2];
        const float4 c1 = crow[lane * 2 + 1];
        float4* o = (float4*)(out + (size_t)batch * 5 * 256 + lane * 8);
        o[0] = c0;
        o[1] = c1;
    }

    __syncthreads();

    // ---------------- Phase 2: h = combined @ W^T + b, SiLU, scatter ---------
    // V_WMMA_F32_16X16X4_F32 layouts (wave32):
    //   A 16x4 : M = lane&15, VGPR0/1 = K {0,1} (lanes 0-15) or {2,3} (lanes 16-31)
    //   B 4x16 : N = lane&15, VGPR0/1 = K {0,1} (lanes 0-15) or {2,3} (lanes 16-31)
    //   C/D    : N = lane&15, VGPR r = row (lane<16 ? r : r+8)
    const int hi = lane >> 4;      // lane-half selects K pair / M-half
    const int nc = lane & 15;

    for (int tile = 0; tile < 2; ++tile) {
        const int nbase = (w * 2 + tile) * 16;                 // 16 tiles cover 256 cols
        const float* wrow = Wm + (size_t)(nbase + nc) * 512 + 2 * hi;  // B[k][n]=W[n][k]
        const float* arow = &sC[nc * LDS_STRIDE + 2 * hi];

        v8f acc = {};
        #pragma unroll 8
        for (int k0 = 0; k0 < 512; k0 += 4) {
            v2f a  = *(const v2f*)&arow[k0];
            v2f bb = *(const v2f*)&wrow[k0];
            acc = __builtin_amdgcn_wmma_f32_16x16x4_f32(
                      /*neg_a=*/false, a, /*neg_b=*/false, bb,
                      /*c_mod=*/(short)0, acc,
                      /*reuse_a=*/false, /*reuse_b=*/false);
        }

        const int   colg = nbase + nc;
        const float bv   = bias[colg];
        #pragma unroll
        for (int r = 0; r < 8; ++r) {
            const int m     = hi ? (r + 8) : r;     // local output row
            const int gm    = blk * 16 + m;         // global n index
            const int batch = gm >> 2;
            const int t     = gm & 3;
            const float x = acc[r] + bv;
            // SiLU via hardware transcendentals: v_exp_f32 + v_rcp_f32
            const float y = x * __builtin_amdgcn_rcpf(1.0f + __expf(-x));
            out[((size_t)batch * 5 + (t + 1)) * 256 + colg] = y;
        }
    }
}

extern "C" void kernel_launch(void* const* d_in, const int* in_sizes, int n_in,
                              void* d_out, int out_size, void* d_ws, size_t ws_size,
                              hipStream_t stream) {
    const int*   nodes    = (const int*)  d_in[0];   // (B,5) int32
    // d_in[1] = mask_row: structurally repeat(arange(N),K) -> not needed
    const int*   mask_col = (const int*)  d_in[2];   // (N*32) int32
    const float* mask_val = (const float*)d_in[3];   // (N*32) f32
    const int*   uniq     = (const int*)  d_in[4];   // (U) int32
    const float* embed    = (const float*)d_in[5];   // (V,256) f32
    const float* Wm       = (const float*)d_in[6];   // (256,512) f32
    const float* bias     = (const float*)d_in[7];   // (256) f32
    float*       out      = (float*)d_out;           // (B,5,256) f32

    const int B      = in_sizes[0] / 5;              // 4096
    const int blocks = B / 4;                        // 16 segments (=4 batches) per block

    hipLaunchKernelGGL(hsagnn_fused_kernel, dim3(blocks), dim3(256), 0, stream,
                       nodes, mask_col, mask_val, uniq, embed, Wm, bias, out);
}